// GOTSim_40080634806646
// MI455X (gfx1250) — compile-verified
//
#include <hip/hip_runtime.h>
#include <hip/hip_bf16.h>

typedef float v2f __attribute__((ext_vector_type(2)));
typedef float v8f __attribute__((ext_vector_type(8)));

#define BATCH 256
#define SNODE 32
#define NLAY  3
#define FIND  512
#define FD    256
#define NNODE (BATCH * SNODE)   // 8192
#define E2    65536             // directed edges per side (2*B*EG)
#define NP    (BATCH * NLAY)    // 768 LAP problems
#define NC    64                // 2S cost matrix dim
#define BIGVAL 99999.0f

#define KC      32              // K-chunk staged in LDS for GEMM
#define WSTRIDE 72              // padded LDS row stride: 2*72 mod 64 == 16 -> halves on disjoint banks

// ---------------------------------------------------------------- utilities
__global__ void fill_f32(float* __restrict__ p, float v, int n) {
    int i = blockIdx.x * blockDim.x + threadIdx.x;
    if (i < n) p[i] = v;
}

__global__ void deg_count(const int* __restrict__ dst, float* __restrict__ deg, int ne) {
    int i = blockIdx.x * blockDim.x + threadIdx.x;
    if (i < ne) atomicAdd(&deg[dst[i]], 1.0f);
}

__global__ void rsqrt_k(float* __restrict__ p, int n) {
    int i = blockIdx.x * blockDim.x + threadIdx.x;
    if (i < n) p[i] = rsqrtf(p[i]);
}

// ---------------------------------------------------------------- fp32 WMMA GEMM
// out[NNODE x FD] = (relu?)(A[NNODE x K]) @ W[K x FD]
// grid = (FD/64, NNODE/128), block = 256 (8 waves); each wave -> 16 rows x 64 cols.
// W chunk (KC x 64) staged in LDS once per block (shared by all 8 waves).
__global__ __launch_bounds__(256)
void gemm_wmma(const float* __restrict__ A, const float* __restrict__ W,
               float* __restrict__ out, int K, int applyRelu) {
    __shared__ __align__(16) float Wl[KC * WSTRIDE];

    int wave = threadIdx.x >> 5;
    int lane = threadIdx.x & 31;
    int m0   = blockIdx.y * 128 + wave * 16;
    int n0   = blockIdx.x * 64;
    int mrow = m0 + (lane & 15);
    int koff = (lane >> 4) << 1;          // 0 or 2
    int ncol = lane & 15;
    const float* Arow = A + (size_t)mrow * K;

    v8f acc0 = {}, acc1 = {}, acc2 = {}, acc3 = {};
    for (int k0 = 0; k0 < K; k0 += KC) {
        // cooperative coalesced load of W[k0..k0+KC-1][n0..n0+63] -> padded LDS
        int idx = threadIdx.x;            // 512 float4s, 2 per thread
        #pragma unroll
        for (int q = 0; q < 2; ++q, idx += 256) {
            int r  = idx >> 4;            // 16 float4 per 64-col row
            int c4 = (idx & 15) << 2;
            const float4 wv = *(const float4*)(W + (size_t)(k0 + r) * FD + n0 + c4);
            *(float4*)(&Wl[r * WSTRIDE + c4]) = wv;
        }
        __syncthreads();

        // prefetch next A chunk while computing this one
        if (k0 + KC < K) __builtin_prefetch(Arow + k0 + KC + koff, 0, 0);

        #pragma unroll
        for (int kk = 0; kk < KC; kk += 4) {
            const float* ap = Arow + k0 + kk + koff;
            v2f a;
            a.x = ap[0]; a.y = ap[1];
            if (applyRelu) { a.x = fmaxf(a.x, 0.f); a.y = fmaxf(a.y, 0.f); }
            const float* b0 = &Wl[(kk + koff) * WSTRIDE + ncol];
            const float* b1 = b0 + WSTRIDE;
            v2f b;
            b.x = b0[0];  b.y = b1[0];
            acc0 = __builtin_amdgcn_wmma_f32_16x16x4_f32(false, a, false, b, (short)0, acc0, false, false);
            b.x = b0[16]; b.y = b1[16];
            acc1 = __builtin_amdgcn_wmma_f32_16x16x4_f32(false, a, false, b, (short)0, acc1, false, false);
            b.x = b0[32]; b.y = b1[32];
            acc2 = __builtin_amdgcn_wmma_f32_16x16x4_f32(false, a, false, b, (short)0, acc2, false, false);
            b.x = b0[48]; b.y = b1[48];
            acc3 = __builtin_amdgcn_wmma_f32_16x16x4_f32(false, a, false, b, (short)0, acc3, false, false);
        }
        __syncthreads();
    }

    int half = lane >> 4;
    #pragma unroll
    for (int r = 0; r < 8; ++r) {
        int row = m0 + r + half * 8;
        float* o = out + (size_t)row * FD + n0 + ncol;
        o[0] = acc0[r]; o[16] = acc1[r]; o[32] = acc2[r]; o[48] = acc3[r];
    }
}

// ---------------------------------------------------------------- GCN aggregation
__global__ void agg_init(const float* __restrict__ lin, const float* __restrict__ dinv,
                         const float* __restrict__ bias, float* __restrict__ feat) {
    int idx = blockIdx.x * blockDim.x + threadIdx.x;
    if (idx >= NNODE * FD) return;
    int node = idx >> 8;                  // FD == 256
    int f    = idx & 255;
    float w = dinv[node] * dinv[node];
    feat[idx] = bias[f] + lin[idx] * w;
}

__global__ void agg_edges(const int* __restrict__ ei, const float* __restrict__ lin,
                          const float* __restrict__ dinv, float* __restrict__ feat, int ne) {
    int idx = blockIdx.x * blockDim.x + threadIdx.x;
    if (idx >= ne * 64) return;
    int e  = idx >> 6;
    int f4 = (idx & 63) << 2;
    int s = ei[e];
    int d = ei[ne + e];
    float w = dinv[s] * dinv[d];
    const float* lp = lin + (size_t)s * FD + f4;
    float* fp = feat + (size_t)d * FD + f4;
    atomicAdd(&fp[0], lp[0] * w);
    atomicAdd(&fp[1], lp[1] * w);
    atomicAdd(&fp[2], lp[2] * w);
    atomicAdd(&fp[3], lp[3] * w);
}

// ---------------------------------------------------------------- sim matrices
// One block per (b,l). Waves 0..3: 16x16 WMMA tiles of main = -q @ c^T.
__global__ __launch_bounds__(128)
void sim_wmma(const float* __restrict__ featq, const float* __restrict__ featc,
              const float* __restrict__ del_p, const float* __restrict__ ins_p,
              float* __restrict__ cost) {
    int prob = blockIdx.x;               // b*NLAY + l
    int b = prob / NLAY, l = prob % NLAY;
    const float* q = featq + ((size_t)l * NNODE + (size_t)b * SNODE) * FD;
    const float* c = featc + ((size_t)l * NNODE + (size_t)b * SNODE) * FD;
    float* Cm = cost + (size_t)prob * NC * NC;

    int wave = threadIdx.x >> 5, lane = threadIdx.x & 31;
    int m0 = (wave >> 1) * 16, n0 = (wave & 1) * 16;
    int koff = (lane >> 4) << 1;
    int mr = m0 + (lane & 15), nr = n0 + (lane & 15);

    v8f acc = {};
    for (int k0 = 0; k0 < FD; k0 += 4) {
        v2f a, bv;
        const float* ap = q + (size_t)mr * FD + k0 + koff;
        a.x = ap[0]; a.y = ap[1];
        const float* bp = c + (size_t)nr * FD + k0 + koff;   // c^T: contiguous in K
        bv.x = bp[0]; bv.y = bp[1];
        acc = __builtin_amdgcn_wmma_f32_16x16x4_f32(false, a, false, bv, (short)0, acc, false, false);
    }
    int half = lane >> 4;
    #pragma unroll
    for (int r = 0; r < 8; ++r) {
        int row = m0 + r + half * 8;
        Cm[row * NC + n0 + (lane & 15)] = -acc[r];
    }

    __shared__ float dv[64];             // dv[0..31] = -(q_i . del_p), dv[32..63] = -(c_i . ins_p)
    int t = threadIdx.x;
    if (t < 64) {
        const float* x = (t < 32) ? (q + (size_t)t * FD) : (c + (size_t)(t - 32) * FD);
        const float* p = (t < 32) ? (del_p + l * FD) : (ins_p + l * FD);
        float s = 0.f;
        for (int k = 0; k < FD; ++k) s += x[k] * p[k];
        dv[t] = -s;
    }
    __syncthreads();

    for (int idx = t; idx < 3 * SNODE * SNODE; idx += blockDim.x) {
        int blk = idx / (SNODE * SNODE);
        int r   = (idx / SNODE) % SNODE;
        int cc  = idx % SNODE;
        int row, col; float val;
        if (blk == 0)      { row = r;          col = SNODE + cc; val = (r == cc) ? dv[r]       : BIGVAL; }
        else if (blk == 1) { row = SNODE + r;  col = cc;         val = (r == cc) ? dv[32 + cc] : BIGVAL; }
        else               { row = SNODE + r;  col = SNODE + cc; val = 0.f; }
        Cm[row * NC + col] = val;
    }
}

// ---------------------------------------------------------------- LAP (Jonker-Volgenant)
__global__ __launch_bounds__(64)
void lap_kernel(const float* __restrict__ cost, float* __restrict__ mcost) {
    __shared__ float cs[NC * NC];
    __shared__ float u[NC], v[NC], shortest[NC], redv[NC];
    __shared__ int   path[NC], SC[NC], col4row[NC], row4col[NC], redi[NC];
    __shared__ float s_min;
    __shared__ int   s_i, s_sink;

    int prob = blockIdx.x;
    int t = threadIdx.x;
    const float* Cg = cost + (size_t)prob * NC * NC;
    for (int k = t; k < NC * NC; k += 64) cs[k] = Cg[k];
    u[t] = 0.f; v[t] = 0.f; col4row[t] = -1; row4col[t] = -1;
    __syncthreads();

    const float INF = 3.0e38f;
    for (int cur = 0; cur < NC; ++cur) {
        shortest[t] = INF; path[t] = -1; SC[t] = 0;
        if (t == 0) { s_i = cur; s_min = 0.f; s_sink = -1; }
        __syncthreads();

        while (true) {
            int i = s_i; float mv = s_min;
            float r = mv + cs[i * NC + t] - u[i] - v[t];
            if (!SC[t] && r < shortest[t]) { shortest[t] = r; path[t] = i; }
            __syncthreads();
            redv[t] = SC[t] ? INF : shortest[t]; redi[t] = t;
            __syncthreads();
            for (int off = 32; off > 0; off >>= 1) {
                if (t < off) {
                    float v2 = redv[t + off]; int i2 = redi[t + off];
                    if (v2 < redv[t] || (v2 == redv[t] && i2 < redi[t])) { redv[t] = v2; redi[t] = i2; }
                }
                __syncthreads();
            }
            if (t == 0) {
                int j = redi[0]; s_min = redv[0]; SC[j] = 1;
                if (row4col[j] < 0) s_sink = j; else s_i = row4col[j];
            }
            __syncthreads();
            if (s_sink >= 0) break;
        }

        float mv = s_min;
        if (t == 0) u[cur] += mv;
        if (SC[t]) {
            int rr = row4col[t];                 // pre-augmentation assignment; rows distinct
            if (rr >= 0) u[rr] += mv - shortest[t];
            v[t] -= mv - shortest[t];
        }
        __syncthreads();
        if (t == 0) {
            int j = s_sink;
            while (true) {
                int i2 = path[j];
                row4col[j] = i2;
                int nj = col4row[i2];
                col4row[i2] = j;
                j = nj;
                if (i2 == cur) break;
            }
        }
        __syncthreads();
    }

    // mcost = sum_i cost[i, col4row[i]]
    redv[t] = cs[t * NC + col4row[t]];
    __syncthreads();
    for (int off = 32; off > 0; off >>= 1) {
        if (t < off) redv[t] += redv[t + off];
        __syncthreads();
    }
    if (t == 0) mcost[prob] = redv[0];
}

// ---------------------------------------------------------------- final score
__global__ void score_kernel(const float* __restrict__ mcost, const float* __restrict__ ot_w,
                             const float* __restrict__ ot_b, float* __restrict__ out) {
    int b = blockIdx.x * blockDim.x + threadIdx.x;
    if (b >= BATCH) return;
    float s = ot_b[0];
    for (int l = 0; l < NLAY; ++l)
        s += (mcost[b * NLAY + l] / (float)SNODE) * ot_w[l];   // 2*mcost/(2S)
    out[b] = 1.f / (1.f + expf(-s));
}

// ---------------------------------------------------------------- launcher
extern "C" void kernel_launch(void* const* d_in, const int* in_sizes, int n_in,
                              void* d_out, int out_size, void* d_ws, size_t ws_size,
                              hipStream_t stream) {
    const float* x_q   = (const float*)d_in[0];
    const float* x_c   = (const float*)d_in[1];
    const float* Wm[3]  = {(const float*)d_in[2], (const float*)d_in[4], (const float*)d_in[6]};
    const float* bia[3] = {(const float*)d_in[3], (const float*)d_in[5], (const float*)d_in[7]};
    const float* del_p = (const float*)d_in[8];
    const float* ins_p = (const float*)d_in[9];
    const float* ot_w  = (const float*)d_in[10];
    const float* ot_b  = (const float*)d_in[11];
    const int*   ei_q  = (const int*)d_in[12];
    const int*   ei_c  = (const int*)d_in[13];

    float* ws = (float*)d_ws;
    size_t off = 0;
    float* dinv_q = ws + off; off += NNODE;
    float* dinv_c = ws + off; off += NNODE;
    float* lin    = ws + off; off += (size_t)NNODE * FD;
    float* featq  = ws + off; off += (size_t)NLAY * NNODE * FD;
    float* featc  = ws + off; off += (size_t)NLAY * NNODE * FD;
    float* cost   = ws + off; off += (size_t)NP * NC * NC;
    float* mcost  = ws + off; off += NP;

    // degrees -> dinv (self-loop included via init to 1.0)
    fill_f32<<<(NNODE + 255) / 256, 256, 0, stream>>>(dinv_q, 1.0f, NNODE);
    fill_f32<<<(NNODE + 255) / 256, 256, 0, stream>>>(dinv_c, 1.0f, NNODE);
    deg_count<<<(E2 + 255) / 256, 256, 0, stream>>>(ei_q + E2, dinv_q, E2);
    deg_count<<<(E2 + 255) / 256, 256, 0, stream>>>(ei_c + E2, dinv_c, E2);
    rsqrt_k<<<(NNODE + 255) / 256, 256, 0, stream>>>(dinv_q, NNODE);
    rsqrt_k<<<(NNODE + 255) / 256, 256, 0, stream>>>(dinv_c, NNODE);

    for (int side = 0; side < 2; ++side) {
        const float* x0   = side ? x_c : x_q;
        const int*   ei   = side ? ei_c : ei_q;
        const float* dinv = side ? dinv_c : dinv_q;
        float*       feat = side ? featc : featq;
        for (int l = 0; l < NLAY; ++l) {
            const float* A = (l == 0) ? x0 : (feat + (size_t)(l - 1) * NNODE * FD);
            int K = (l == 0) ? FIND : FD;
            dim3 grid(FD / 64, NNODE / 128);
            gemm_wmma<<<grid, 256, 0, stream>>>(A, Wm[l], lin, K, l > 0);
            float* fl = feat + (size_t)l * NNODE * FD;
            agg_init<<<(NNODE * FD + 255) / 256, 256, 0, stream>>>(lin, dinv, bia[l], fl);
            agg_edges<<<(E2 * 64 + 255) / 256, 256, 0, stream>>>(ei, lin, dinv, fl, E2);
        }
    }

    sim_wmma<<<NP, 128, 0, stream>>>(featq, featc, del_p, ins_p, cost);
    lap_kernel<<<NP, 64, 0, stream>>>(cost, mcost);
    score_kernel<<<1, 256, 0, stream>>>(mcost, ot_w, ot_b, (float*)d_out);
}